// GNN_46273977647810
// MI455X (gfx1250) — compile-verified
//
#include <hip/hip_runtime.h>
#include <math.h>

typedef float v2f __attribute__((ext_vector_type(2)));
typedef float v4f __attribute__((ext_vector_type(4)));
typedef float v8f __attribute__((ext_vector_type(8)));

#define DIM  256   // 2^8 amplitudes
#define LDSS 260   // padded LDS row stride (260 % 64 == 4 -> conflict-free)

// ---------------------------------------------------------------------------
// Phase 1: recover src/dst indices from the dense one-hot incidence matrices.
// Ri[n,c]==1 at n==dst[c]; Ro[n,c]==1 at n==src[c]. Exactly one hit/column,
// so plain stores suffice. This 512MB nontemporal read is the roofline floor.
// ---------------------------------------------------------------------------
__global__ void k_extract(const float* __restrict__ Ri, const float* __restrict__ Ro,
                          int* __restrict__ dstI, int* __restrict__ srcI,
                          int total4, int E) {
  int stride = gridDim.x * blockDim.x;
  for (int i = blockIdx.x * blockDim.x + threadIdx.x; i < total4; i += stride) {
    v4f ri = __builtin_nontemporal_load(((const v4f*)Ri) + i);
    v4f ro = __builtin_nontemporal_load(((const v4f*)Ro) + i);
    bool anyi = (ri.x != 0.f) | (ri.y != 0.f) | (ri.z != 0.f) | (ri.w != 0.f);
    bool anyo = (ro.x != 0.f) | (ro.y != 0.f) | (ro.z != 0.f) | (ro.w != 0.f);
    if (anyi | anyo) {
      int base = i << 2;
      int n = base / E;        // rows are E-aligned; E % 4 == 0 so no row crossing
      int c = base - n * E;
      if (ri.x != 0.f) dstI[c + 0] = n;
      if (ri.y != 0.f) dstI[c + 1] = n;
      if (ri.z != 0.f) dstI[c + 2] = n;
      if (ri.w != 0.f) dstI[c + 3] = n;
      if (ro.x != 0.f) srcI[c + 0] = n;
      if (ro.y != 0.f) srcI[c + 1] = n;
      if (ro.z != 0.f) srcI[c + 2] = n;
      if (ro.w != 0.f) srcI[c + 3] = n;
    }
  }
}

// H = [X @ kernel, X]  -> [N,4]
__global__ void k_H(const float* __restrict__ X, const float* __restrict__ kern,
                    float* __restrict__ H, int N) {
  int n = blockIdx.x * blockDim.x + threadIdx.x;
  if (n >= N) return;
  float x0 = X[3*n], x1 = X[3*n+1], x2 = X[3*n+2];
  H[4*n+0] = x0*kern[0] + x1*kern[1] + x2*kern[2];
  H[4*n+1] = x0; H[4*n+2] = x1; H[4*n+3] = x2;
}

// ---------------------------------------------------------------------------
// Phase 2: build U(theta) = product of the fixed gate sequence, one wave per
// column. Amplitude index k = lane*8 + r (wire w <-> bit 7-w). Gates on bits
// 0..2 are in-register; bits 3..7 are lane bits handled with __shfl_xor.
// ---------------------------------------------------------------------------
template<int W>
__device__ __forceinline__ void ry_g(float (&st)[8], float c, float s, int lane) {
  constexpr int mask = 1 << (7 - W);
  if constexpr (mask < 8) {
#pragma unroll
    for (int r = 0; r < 8; ++r) {
      if ((r & mask) == 0) {
        float v0 = st[r], v1 = st[r | mask];
        st[r]        = c * v0 - s * v1;
        st[r | mask] = s * v0 + c * v1;
      }
    }
  } else {
    constexpr int lm = mask >> 3;
#pragma unroll
    for (int r = 0; r < 8; ++r) {
      float p = __shfl_xor(st[r], lm, 32);
      st[r] = (lane & lm) ? (s * p + c * st[r]) : (c * st[r] - s * p);
    }
  }
}

template<int C, int T>
__device__ __forceinline__ void cx_g(float (&st)[8], int lane) {
  constexpr int cm = 1 << (7 - C);
  constexpr int tm = 1 << (7 - T);
  float ns[8];
#pragma unroll
  for (int r = 0; r < 8; ++r) {
    float pv;
    if constexpr (tm < 8) pv = st[r ^ tm];
    else                  pv = __shfl_xor(st[r], tm >> 3, 32);
    bool cb;
    if constexpr (cm < 8) cb = (r & cm) != 0;
    else                  cb = (lane & (cm >> 3)) != 0;
    ns[r] = cb ? pv : st[r];
  }
#pragma unroll
  for (int r = 0; r < 8; ++r) st[r] = ns[r];
}

__global__ void __launch_bounds__(256) k_buildU(const float* __restrict__ t0,
                                                const float* __restrict__ t1,
                                                const float* __restrict__ t2,
                                                float* __restrict__ U) {
  int gid  = blockIdx.x * blockDim.x + threadIdx.x;
  int lane = gid & 31;
  int wv   = gid >> 5;
  int j    = wv & 255;       // column of U
  int u    = wv >> 8;        // which theta set (0..2)
  const float* th = (u == 0) ? t0 : (u == 1) ? t1 : t2;
  float cg[15], sg[15];
#pragma unroll
  for (int i = 0; i < 15; ++i) { float a = th[i] * 0.5f; cg[i] = cosf(a); sg[i] = sinf(a); }
  float st[8];
#pragma unroll
  for (int r = 0; r < 8; ++r) st[r] = ((lane * 8 + r) == j) ? 1.f : 0.f;

  ry_g<0>(st,cg[0],sg[0],lane);  ry_g<1>(st,cg[1],sg[1],lane);  cx_g<0,1>(st,lane);
  ry_g<2>(st,cg[2],sg[2],lane);  ry_g<3>(st,cg[3],sg[3],lane);  cx_g<3,2>(st,lane);
  ry_g<4>(st,cg[4],sg[4],lane);  ry_g<5>(st,cg[5],sg[5],lane);  cx_g<4,5>(st,lane);
  ry_g<6>(st,cg[6],sg[6],lane);  ry_g<7>(st,cg[7],sg[7],lane);  cx_g<7,6>(st,lane);
  ry_g<1>(st,cg[8],sg[8],lane);  ry_g<2>(st,cg[9],sg[9],lane);  cx_g<1,2>(st,lane);
  ry_g<5>(st,cg[10],sg[10],lane); ry_g<6>(st,cg[11],sg[11],lane); cx_g<6,5>(st,lane);
  ry_g<2>(st,cg[12],sg[12],lane); ry_g<5>(st,cg[13],sg[13],lane); cx_g<2,5>(st,lane);
  ry_g<5>(st,cg[14],sg[14],lane);

  float* Uo = U + (size_t)u * DIM * DIM;   // row-major: Uo[k*256 + j]
#pragma unroll
  for (int r = 0; r < 8; ++r) Uo[(lane * 8 + r) * DIM + j] = st[r];
}

// feats for EdgeNet0: [H[src], H[dst]] per edge
__global__ void k_feats_e0(const float* __restrict__ H, const int* __restrict__ srcI,
                           const int* __restrict__ dstI, float* __restrict__ feats, int E) {
  int c = blockIdx.x * blockDim.x + threadIdx.x;
  if (c >= E) return;
  int s = srcI[c], d = dstI[c];
#pragma unroll
  for (int j = 0; j < 4; ++j) {
    feats[c*8 + j]     = H[s*4 + j];
    feats[c*8 + 4 + j] = H[d*4 + j];
  }
}

// mi[d] += e*H[src]; mo[s] += e*H[dst]  (mimo layout: [N][mi0..3, mo0..3])
__global__ void k_nodeagg(const float* __restrict__ H, const float* __restrict__ e,
                          const int* __restrict__ srcI, const int* __restrict__ dstI,
                          float* __restrict__ mimo, int E) {
  int c = blockIdx.x * blockDim.x + threadIdx.x;
  if (c >= E) return;
  float w = e[c];
  int s = srcI[c], d = dstI[c];
#pragma unroll
  for (int j = 0; j < 4; ++j) {
    atomicAdd(&mimo[d*8 + j],     w * H[s*4 + j]);
    atomicAdd(&mimo[s*8 + 4 + j], w * H[d*4 + j]);
  }
}

// feats for EdgeNet1: H2[n] = {h[n], X[n]};  [H2[src], H2[dst]]
__global__ void k_feats_e1(const float* __restrict__ X, const float* __restrict__ h,
                           const int* __restrict__ srcI, const int* __restrict__ dstI,
                           float* __restrict__ feats, int E) {
  int c = blockIdx.x * blockDim.x + threadIdx.x;
  if (c >= E) return;
  int s = srcI[c], d = dstI[c];
  feats[c*8+0] = h[s];
  feats[c*8+1] = X[3*s]; feats[c*8+2] = X[3*s+1]; feats[c*8+3] = X[3*s+2];
  feats[c*8+4] = h[d];
  feats[c*8+5] = X[3*d]; feats[c*8+6] = X[3*d+1]; feats[c*8+7] = X[3*d+2];
}

// ---------------------------------------------------------------------------
// Phase 3: out[b] = sum_{n: bit2(n)==1} ( sum_k S0[b,k] * U[n,k] )^2
// 16-row tile per block; S0 tile in LDS; K=256 loop of v_wmma_f32_16x16x4_f32.
// Fragment layouts per ISA 7.12.2: A VGPR0 = K{0|2}, VGPR1 = K{1|3} (halves);
// B symmetric over N-lanes; C/D VGPR r = rows r / r+8.
// ---------------------------------------------------------------------------
__global__ void __launch_bounds__(256) k_gemm_ttn(const float* __restrict__ feats,
                                                  const float* __restrict__ U,
                                                  float* __restrict__ out, int B) {
  __shared__ float cs[16][8][2];
  __shared__ float s0[16 * LDSS];
  __shared__ float red[16];
  const int tid  = threadIdx.x;
  const int row0 = blockIdx.x * 16;

  if (tid < 16) red[tid] = 0.f;
  if (tid < 128) {
    int m = tid >> 3, w = tid & 7;
    int b = row0 + m;
    float f = (b < B) ? feats[b*8 + w] * 0.5f : 0.f;
    cs[m][w][0] = cosf(f);
    cs[m][w][1] = sinf(f);
  }
  __syncthreads();
  // product state: S0[m][k] = prod_w cs[m][w][bit(7-w) of k]
  for (int idx = tid; idx < 16 * DIM; idx += 256) {
    int m = idx >> 8, k = idx & 255;
    float p = cs[m][0][(k>>7)&1];
    p *= cs[m][1][(k>>6)&1];
    p *= cs[m][2][(k>>5)&1];
    p *= cs[m][3][(k>>4)&1];
    p *= cs[m][4][(k>>3)&1];
    p *= cs[m][5][(k>>2)&1];
    p *= cs[m][6][(k>>1)&1];
    p *= cs[m][7][ k     &1];
    s0[m * LDSS + k] = p;
  }
  __syncthreads();

  const int wave = tid >> 5;
  const int lane = tid & 31;
  const int half = lane >> 4;
  const int ml   = lane & 15;

  for (int ct = wave; ct < 16; ct += 8) {      // uniform per wave -> EXEC all 1s
    const int n = ct * 16 + ml;                // amplitude (output) index
    const float* urow = U + n * DIM;           // B[k][n] = U[n][k]
    const float* arow = s0 + ml * LDSS + 2 * half;
    v8f acc = {0.f, 0.f, 0.f, 0.f, 0.f, 0.f, 0.f, 0.f};
#pragma unroll 4
    for (int kb = 0; kb < 64; ++kb) {
      v2f a, bf;
      a.x = arow[4*kb];                        // S0[ml][4kb + 2*half + 0]
      a.y = arow[4*kb + 1];                    // S0[ml][4kb + 2*half + 1]
      bf  = *(const v2f*)(urow + 4*kb + 2*half);
      acc = __builtin_amdgcn_wmma_f32_16x16x4_f32(false, a, false, bf,
                                                  (short)0, acc, false, false);
    }
    const bool msk = ((n >> 2) & 1) != 0;      // measured wire 5 -> bit 2
    float p[8];
#pragma unroll
    for (int r = 0; r < 8; ++r) p[r] = msk ? acc[r] * acc[r] : 0.f;
#pragma unroll
    for (int d = 1; d < 16; d <<= 1) {
#pragma unroll
      for (int r = 0; r < 8; ++r) p[r] += __shfl_xor(p[r], d, 32);
    }
    if (ml == 0) {
#pragma unroll
      for (int r = 0; r < 8; ++r) atomicAdd(&red[half * 8 + r], p[r]);
    }
  }
  __syncthreads();
  if (tid < 16 && row0 + tid < B) out[row0 + tid] = red[tid];
}

// ---------------------------------------------------------------------------
extern "C" void kernel_launch(void* const* d_in, const int* in_sizes, int n_in,
                              void* d_out, int out_size, void* d_ws, size_t ws_size,
                              hipStream_t stream) {
  (void)n_in; (void)out_size; (void)ws_size;
  const float* X    = (const float*)d_in[0];
  const float* Ri   = (const float*)d_in[1];
  const float* Ro   = (const float*)d_in[2];
  const float* kern = (const float*)d_in[3];
  const float* te0  = (const float*)d_in[4];
  const float* tn   = (const float*)d_in[5];
  const float* te1  = (const float*)d_in[6];
  const int N = in_sizes[0] / 3;
  const int E = in_sizes[1] / N;

  char* p = (char*)d_ws;
  int*   srcI  = (int*)p;    p += (size_t)E * 4;
  int*   dstI  = (int*)p;    p += (size_t)E * 4;
  float* H     = (float*)p;  p += (size_t)N * 4 * 4;
  float* U     = (float*)p;  p += (size_t)3 * DIM * DIM * 4;
  float* ev    = (float*)p;  p += (size_t)E * 4;
  float* hv    = (float*)p;  p += (size_t)N * 4;
  float* mimo  = (float*)p;  p += (size_t)N * 8 * 4;
  float* feats = (float*)p;  p += (size_t)E * 8 * 4;

  hipMemsetAsync(mimo, 0, (size_t)N * 8 * 4, stream);
  k_extract<<<4096, 256, 0, stream>>>(Ri, Ro, dstI, srcI, (N / 4) * E, E);
  k_H<<<(N + 255) / 256, 256, 0, stream>>>(X, kern, H, N);
  k_buildU<<<96, 256, 0, stream>>>(te0, tn, te1, U);        // 768 waves = 3*256 cols
  k_feats_e0<<<(E + 255) / 256, 256, 0, stream>>>(H, srcI, dstI, feats, E);
  k_gemm_ttn<<<(E + 15) / 16, 256, 0, stream>>>(feats, U, ev, E);
  k_nodeagg<<<(E + 255) / 256, 256, 0, stream>>>(H, ev, srcI, dstI, mimo, E);
  k_gemm_ttn<<<(N + 15) / 16, 256, 0, stream>>>(mimo, U + DIM * DIM, hv, N);
  k_feats_e1<<<(E + 255) / 256, 256, 0, stream>>>(X, hv, srcI, dstI, feats, E);
  k_gemm_ttn<<<(E + 15) / 16, 256, 0, stream>>>(feats, U + 2 * DIM * DIM,
                                                (float*)d_out, E);
}